// Row_Averaging_78142634983969
// MI455X (gfx1250) — compile-verified
//
#include <hip/hip_runtime.h>

typedef __attribute__((ext_vector_type(2))) float v2f;
typedef __attribute__((ext_vector_type(8))) float v8f;

#if defined(__has_builtin)
#  if __has_builtin(__builtin_amdgcn_wmma_f32_16x16x4_f32)
#    define HAVE_WMMA_F32X4 1
#  else
#    define HAVE_WMMA_F32X4 0
#  endif
#else
#  define HAVE_WMMA_F32X4 0
#endif

#define H28 28
#define WAVES_PER_BLOCK 8
#define ROWS_PER_WAVE 32
#define BLOCK_THREADS (WAVES_PER_BLOCK * 32)

// One wave processes 32 rows:
//   Phase 1: two WMMA f32 16x16x4 batches compute row means (B-operand = 1/28,
//            7 K-chunks cover the 28 columns). D replicates each row's mean
//            across all 16 columns; lanes 0 and 16 spill 8 means each to LDS.
//   Phase 2: lane-per-row masked streaming copy: out = mask ? mean : x.
__global__ __launch_bounds__(BLOCK_THREADS)
void Row_Averaging_78142634983969_kernel(const float* __restrict__ x,
                                         const int*   __restrict__ t,
                                         const int*   __restrict__ rows_t,
                                         float*       __restrict__ out,
                                         int nrows) {
    __shared__ int   pos[H28];
    __shared__ float means[WAVES_PER_BLOCK * ROWS_PER_WAVE];

    const int tid = threadIdx.x;
    if (tid < H28) pos[rows_t[tid]] = tid;
    __syncthreads();

    const int lane = tid & 31;
    const int wave = tid >> 5;
    const int rowBase = (blockIdx.x * WAVES_PER_BLOCK + wave) * ROWS_PER_WAVE;

#if HAVE_WMMA_F32X4
    // ---- Phase 1: WMMA row-mean reduction -------------------------------
    // A (16x4 f32) layout: lanes 0-15 -> {K0,K1}, lanes 16-31 -> {K2,K3}.
    const int mLocal = lane & 15;            // row-in-batch carried by this lane
    const int koff   = (lane >> 4) << 1;     // 0 (low half) or 2 (high half)
    v2f bmat;
    bmat[0] = 1.0f / 28.0f;                  // B is constant -> its K<->lane
    bmat[1] = 1.0f / 28.0f;                  // mapping is irrelevant

    #pragma unroll
    for (int batch = 0; batch < 2; ++batch) {
        int row = rowBase + batch * 16 + mLocal;
        int srow = (row < nrows) ? row : 0;  // clamp: keep wave convergent
        const float* rp = x + (size_t)srow * H28 + koff;
        v8f acc = {};
        #pragma unroll
        for (int k = 0; k < 7; ++k) {
            v2f a = *(const v2f*)(rp + 4 * k);   // 8B-aligned (row pitch 112B)
            acc = __builtin_amdgcn_wmma_f32_16x16x4_f32(
                false, a, false, bmat, (short)0, acc, false, false);
        }
        // D layout: VGPR r = row r (lanes 0-15) / row 8+r (lanes 16-31),
        // every column identical -> lanes 0 and 16 hold all 8 means each.
        if ((lane & 15) == 0) {
            int off = wave * ROWS_PER_WAVE + batch * 16 + ((lane >> 4) << 3);
            #pragma unroll
            for (int r = 0; r < 8; ++r) means[off + r] = acc[r];
        }
    }
#endif
    __syncthreads();

    // ---- Phase 2: masked streaming copy (hits WGP$ for the re-read) -----
    const int row = rowBase + lane;
    if (row < nrows) {
        const int b = row / H28;
        const int h = row - b * H28;
        const bool msk = pos[h] < t[b];

        const float4* xr = (const float4*)(x + (size_t)row * H28);
        float4*       orw = (float4*)(out + (size_t)row * H28);

        float4 v[7];
        #pragma unroll
        for (int k = 0; k < 7; ++k) v[k] = xr[k];

#if HAVE_WMMA_F32X4
        const float mean = means[wave * ROWS_PER_WAVE + lane];
#else
        float s = 0.0f;
        #pragma unroll
        for (int k = 0; k < 7; ++k) s += v[k].x + v[k].y + v[k].z + v[k].w;
        const float mean = s * (1.0f / 28.0f);
#endif
        #pragma unroll
        for (int k = 0; k < 7; ++k) {
            float4 o;
            o.x = msk ? mean : v[k].x;
            o.y = msk ? mean : v[k].y;
            o.z = msk ? mean : v[k].z;
            o.w = msk ? mean : v[k].w;
            orw[k] = o;
        }
    }
}

extern "C" void kernel_launch(void* const* d_in, const int* in_sizes, int n_in,
                              void* d_out, int out_size, void* d_ws, size_t ws_size,
                              hipStream_t stream) {
    const float* x      = (const float*)d_in[0];  // [B,1,28,28] f32
    const int*   t      = (const int*)d_in[1];    // [B] int32
    const int*   rows_t = (const int*)d_in[2];    // [28] int32 permutation
    float*       out    = (float*)d_out;

    const int B     = in_sizes[1];
    const int nrows = B * H28;
    const int rowsPerBlock = WAVES_PER_BLOCK * ROWS_PER_WAVE;  // 256
    const int grid  = (nrows + rowsPerBlock - 1) / rowsPerBlock;

    Row_Averaging_78142634983969_kernel<<<grid, BLOCK_THREADS, 0, stream>>>(
        x, t, rows_t, out, nrows);
}